// DepthLSSTransform_Seg_debug_42107859370546
// MI455X (gfx1250) — compile-verified
//
#include <hip/hip_runtime.h>

typedef __attribute__((ext_vector_type(2))) float v2f;
typedef __attribute__((ext_vector_type(8))) float v8f;

// ---------------------------------------------------------------- constants
#define BN6      6
#define NCAM     6
#define INCH     256
#define FH_      32
#define FW_      88
#define HW_FEAT  (FH_ * FW_)     // 2816
#define IH_      256
#define IW_      704
#define DBINS    59
#define CCH      80
#define NXG      360
#define NYG      360
#define NZG      1

// Workspace layout (float offsets); regions reused across pipeline phases.
static constexpr size_t OFF_DT1    = 0;          // 8,650,752  (dead before BEV phase)
static constexpr size_t OFF_BEV    = 0;          // 10,368,000 (reuses DT1 region)
static constexpr size_t OFF_DT2    = 10368000;   // 2,162,688  (dead before BEV2)
static constexpr size_t OFF_BEV2   = 10368000;   // 2,592,000  (reuses DT2 region)
static constexpr size_t OFF_CONCAT = 12960000;   // 5,406,720  (dead after dn1)
static constexpr size_t OFF_BEV1   = 12960000;   // 10,368,000 (reuses CONCAT region)
static constexpr size_t OFF_DN1    = 23328000;   // 4,325,376
static constexpr size_t OFF_DN2    = 27653376;   // 4,325,376
static constexpr size_t OFF_DN3    = 31978752;   // 2,348,544
static constexpr size_t OFF_DEPTH  = 34327296;   // 996,864
static constexpr size_t OFF_GEOM   = 35324160;   // 144

// ---------------------------------------------------------------- helpers
static __device__ __forceinline__ void inv3(const float* A, float* o) {
  float a = A[0], b = A[1], c = A[2];
  float d = A[3], e = A[4], f = A[5];
  float g = A[6], h = A[7], i = A[8];
  float det = a * (e * i - f * h) - b * (d * i - f * g) + c * (d * h - e * g);
  float r = 1.0f / det;
  o[0] = (e * i - f * h) * r; o[1] = (c * h - b * i) * r; o[2] = (b * f - c * e) * r;
  o[3] = (f * g - d * i) * r; o[4] = (a * i - c * g) * r; o[5] = (c * d - a * f) * r;
  o[6] = (d * h - e * g) * r; o[7] = (b * g - a * h) * r; o[8] = (a * e - b * d) * r;
}

static __device__ __forceinline__ void mm3(const float* A, const float* B, float* C) {
  for (int r = 0; r < 3; ++r)
    for (int c = 0; c < 3; ++c)
      C[r * 3 + c] = A[r * 3 + 0] * B[0 * 3 + c] + A[r * 3 + 1] * B[1 * 3 + c] + A[r * 3 + 2] * B[2 * 3 + c];
}

// ------------------------------------------------ direct conv (+BN+ReLU) for depth stem
__global__ __launch_bounds__(256) void conv_direct_kernel(
    const float* __restrict__ in, const float* __restrict__ w,
    const float* __restrict__ scale, const float* __restrict__ shift,
    float* __restrict__ out,
    int BN, int Cin, int Cout, int H, int W, int OH, int OW,
    int KH, int KW, int stride, int pad, int outChTotal, int outChOff)
{
  int idx = blockIdx.x * blockDim.x + threadIdx.x;
  int total = BN * Cout * OH * OW;
  if (idx >= total) return;
  int ow = idx % OW; int t = idx / OW;
  int oh = t % OH;   t /= OH;
  int oc = t % Cout; int bn = t / Cout;
  const float* wb  = w  + (size_t)oc * Cin * KH * KW;
  const float* inb = in + (size_t)bn * Cin * H * W;
  float acc = 0.0f;
  for (int ic = 0; ic < Cin; ++ic)
    for (int kh = 0; kh < KH; ++kh) {
      int ih = oh * stride + kh - pad;
      if (ih < 0 || ih >= H) continue;
      for (int kw = 0; kw < KW; ++kw) {
        int iw = ow * stride + kw - pad;
        if (iw < 0 || iw >= W) continue;
        acc += inb[((size_t)ic * H + ih) * W + iw] * wb[(ic * KH + kh) * KW + kw];
      }
    }
  float v = fmaxf(acc * scale[oc] + shift[oc], 0.0f);
  out[(((size_t)bn * outChTotal + outChOff + oc) * OH + oh) * OW + ow] = v;
}

// ------------------------------------------------ implicit-GEMM conv via V_WMMA_F32_16X16X4_F32
// One wave computes a 16(Cout) x 64(pixel) tile: 4 x v8f accumulators, A fragment reused 4x.
// GEMM-K is walked as (kh, kw, ic): kh/kw are outer loops (predicates + pixel offsets hoisted),
// the hot loop strides ic only -> no integer division per step.
// Per (kh,kw) the 16 x Cin weight segment is staged in LDS (row stride Cin+2: even for
// 8B-aligned ds_load_b64 fragments, 2-bank stagger across the 16 rows -> conflict-free).
// The NEXT tap's weight segment is prefetched (global_prefetch_b8) so its staging loads
// hit WGP$/L2 while the current tap's WMMAs execute.
// All lane guards are selects; EXEC stays all-ones around the WMMAs (wave32, 1 wave/WG).
__global__ __launch_bounds__(32) void conv_wmma_kernel(
    const float* __restrict__ in, const float* __restrict__ w,
    const float* __restrict__ scale, const float* __restrict__ shift,
    float* __restrict__ out,
    int BN, int Cin, int Cout, int H, int W, int OH, int OW,
    int KH, int KW, int stride, int pad, int relu)
{
  extern __shared__ float smem[];          // 16 * (Cin + 2) floats
  const int lane = threadIdx.x;
  const int half = lane >> 4;
  const int l16  = lane & 15;
  const int Npix = BN * OH * OW;
  const int KHW  = KH * KW;
  const int K    = Cin * KHW;
  const int CinP = Cin + 2;
  const size_t HWin   = (size_t)H * W;
  const size_t chImg  = (size_t)Cin * HWin;

  // ---- per-subtile pixel decode (4 x 16 pixels) ----
  const int nbase = blockIdx.x * 64;
  bool nv[4]; int ihb[4], iwb[4]; const float* inp[4];
  int bnA[4], ohA[4], owA[4];
#pragma unroll
  for (int j = 0; j < 4; ++j) {
    int n = nbase + j * 16 + l16;
    nv[j] = (n < Npix);
    int t = nv[j] ? n : 0;
    int ow = t % OW; t /= OW;
    int oh = t % OH;
    int bn = t / OH;
    bnA[j] = bn; ohA[j] = oh; owA[j] = ow;
    ihb[j] = oh * stride - pad;
    iwb[j] = ow * stride - pad;
    inp[j] = in + (size_t)bn * chImg;
  }

  // ---- weight staging source for this wave's Cout tile ----
  const int srow  = lane >> 1;                       // 0..15, two lanes per row
  const int mrow  = blockIdx.y * 16 + srow;
  const bool rok  = (mrow < Cout);
  const float* wsrc = w + (size_t)(rok ? mrow : 0) * K;
  const float rmask = rok ? 1.0f : 0.0f;

  v8f acc[4];
#pragma unroll
  for (int j = 0; j < 4; ++j) acc[j] = (v8f){};

  for (int kh = 0; kh < KH; ++kh) {
    for (int kw = 0; kw < KW; ++kw) {
      const int khkw = kh * KW + kw;
      // ---- stage A segment (16 rows x Cin) into LDS ----
      __syncthreads();                               // 1-wave WG: ~S_NOP, orders LDS reuse
      {
        const float* ws2 = wsrc + khkw;
        for (int ic = (lane & 1); ic < Cin; ic += 2)
          smem[srow * CinP + ic] = ws2[(size_t)ic * KHW] * rmask;
      }
      __syncthreads();

      // ---- prefetch next tap's weight segment (overlaps the WMMA loop below) ----
      if (khkw + 1 < KHW) {
        const float* ws3 = wsrc + khkw + 1;          // this lane's staging row, next tap
        for (int ic = (lane & 1); ic < Cin; ic += 16)
          __builtin_prefetch(&ws3[(size_t)ic * KHW], 0, 1);  // global_prefetch_b8
      }

      // ---- per-subtile padding predicate + pixel offset for this tap ----
      bool ok[4]; int poff[4];
#pragma unroll
      for (int j = 0; j < 4; ++j) {
        int ih = ihb[j] + kh, iw = iwb[j] + kw;
        bool o = nv[j] && ((unsigned)ih < (unsigned)H) && ((unsigned)iw < (unsigned)W);
        ok[j] = o;
        poff[j] = o ? (ih * W + iw) : 0;
      }

      // ---- hot loop: stride ic only ----
      for (int ic0 = 0; ic0 < Cin; ic0 += 4) {
        const int icl = ic0 + 2 * half;              // this lane's first K of the fragment
        v2f a = *(const v2f*)&smem[l16 * CinP + icl];  // ds_load_b64, conflict-free
#pragma unroll
        for (int j = 0; j < 4; ++j) {
          const float* p = inp[j] + (size_t)icl * HWin + poff[j];
          float b0 = p[0];
          float b1 = p[HWin];
          v2f b;
          b.x = ok[j] ? b0 : 0.0f;
          b.y = ok[j] ? b1 : 0.0f;
          acc[j] = __builtin_amdgcn_wmma_f32_16x16x4_f32(
              /*neg_a=*/false, a, /*neg_b=*/false, b,
              /*c_mod=*/(short)0, acc[j], /*reuse_a=*/false, /*reuse_b=*/false);
        }
      }
    }
  }

  // ---- epilogue: BN/bias + ReLU, D layout: VGPR i -> row i + 8*half, col l16 ----
  const bool useScale = (scale != nullptr);
  const bool useShift = (shift != nullptr);
#pragma unroll
  for (int j = 0; j < 4; ++j) {
    if (!nv[j]) continue;
    for (int i = 0; i < 8; ++i) {
      int mm = blockIdx.y * 16 + half * 8 + i;
      if (mm < Cout) {
        float v = acc[j][i];
        if (useScale)      v = v * scale[mm] + shift[mm];
        else if (useShift) v = v + shift[mm];
        if (relu) v = fmaxf(v, 0.0f);
        out[(((size_t)bnA[j] * Cout + mm) * OH + ohA[j]) * OW + owA[j]] = v;
      }
    }
  }
}

// ------------------------------------------------ copy x into concat channels [64,320)
__global__ __launch_bounds__(256) void copy_channels_kernel(
    const float* __restrict__ x, float* __restrict__ dst,
    int total, int Cx, int HW, int chTotal, int chOff)
{
  int idx = blockIdx.x * blockDim.x + threadIdx.x;
  if (idx >= total) return;
  int pix = idx % HW; int t = idx / HW;
  int c = t % Cx; int bn = t / Cx;
  dst[((size_t)bn * chTotal + chOff + c) * HW + pix] = x[idx];
}

// ------------------------------------------------ softmax over the 59 depth channels
__global__ __launch_bounds__(256) void softmax_depth_kernel(
    const float* __restrict__ h, float* __restrict__ depth,
    int BN, int D, int HW, int chTotal)
{
  int idx = blockIdx.x * blockDim.x + threadIdx.x;
  int total = BN * HW;
  if (idx >= total) return;
  int pix = idx % HW; int bn = idx / HW;
  const float* base = h + (size_t)bn * chTotal * HW + pix;
  float mx = -3.4e38f;
  for (int d = 0; d < D; ++d) mx = fmaxf(mx, base[(size_t)d * HW]);
  float s = 0.0f;
  for (int d = 0; d < D; ++d) s += expf(base[(size_t)d * HW] - mx);
  float inv = 1.0f / s;
  for (int d = 0; d < D; ++d)
    depth[((size_t)bn * D + d) * HW + pix] = expf(base[(size_t)d * HW] - mx) * inv;
}

// ------------------------------------------------ per-camera geometry prep (3x3 inverses)
__global__ void geom_prep_kernel(
    const float* __restrict__ intrins, const float* __restrict__ post_rots,
    const float* __restrict__ post_trans, const float* __restrict__ c2l_rots,
    const float* __restrict__ c2l_trans, const float* __restrict__ extra_rots,
    const float* __restrict__ extra_trans, float* __restrict__ geom, int BN, int Nn)
{
  int bn = blockIdx.x * blockDim.x + threadIdx.x;
  if (bn >= BN) return;
  int b = bn / Nn;
  float invP[9]; inv3(post_rots + (size_t)bn * 9, invP);
  float invI[9]; inv3(intrins  + (size_t)bn * 9, invI);
  float comb[9]; mm3(c2l_rots + (size_t)bn * 9, invI, comb);
  float M3[9];   mm3(extra_rots + (size_t)b * 9, comb, M3);
  const float* er = extra_rots + (size_t)b * 9;
  const float* ct = c2l_trans + (size_t)bn * 3;
  const float* et = extra_trans + (size_t)b * 3;
  float* o = geom + (size_t)bn * 24;
  for (int i = 0; i < 9; ++i) o[i] = invP[i];
  for (int i = 0; i < 9; ++i) o[9 + i] = M3[i];
  for (int r = 0; r < 3; ++r)
    o[18 + r] = er[r * 3 + 0] * ct[0] + er[r * 3 + 1] * ct[1] + er[r * 3 + 2] * ct[2] + et[r];
  for (int i = 0; i < 3; ++i) o[21 + i] = post_trans[(size_t)bn * 3 + i];
}

__global__ __launch_bounds__(256) void zero_kernel(float* __restrict__ p, int n) {
  int i = blockIdx.x * blockDim.x + threadIdx.x;
  if (i < n) p[i] = 0.0f;
}

// ------------------------------------------------ frustum unproject + voxel scatter-sum
__global__ __launch_bounds__(256) void scatter_kernel(
    const float* __restrict__ dn3, const float* __restrict__ depth,
    const float* __restrict__ geom, float* __restrict__ bev,
    int BN, int Nn, int D, int FH, int FW, int Ctot, int Coff, int C,
    int NX, int NY, int NZ)
{
  int p = blockIdx.x * blockDim.x + threadIdx.x;
  int HW = FH * FW;
  int total = BN * D * HW;
  if (p >= total) return;
  int w = p % FW; int t = p / FW;
  int h = t % FH; t /= FH;
  int dd = t % D; int bn = t / D;

  const float* g = geom + (size_t)bn * 24;
  float xs = (float)w * ((float)(IW_ - 1) / (float)(FW_ - 1));
  float ys = (float)h * ((float)(IH_ - 1) / (float)(FH_ - 1));
  float zd = 1.0f + (float)dd;
  float px = xs - g[21], py = ys - g[22], pz = zd - g[23];
  float qx = g[0] * px + g[1] * py + g[2] * pz;
  float qy = g[3] * px + g[4] * py + g[5] * pz;
  float qz = g[6] * px + g[7] * py + g[8] * pz;
  qx *= qz; qy *= qz;
  float gx = g[9]  * qx + g[10] * qy + g[11] * qz + g[18];
  float gy = g[12] * qx + g[13] * qy + g[14] * qz + g[19];
  float gz = g[15] * qx + g[16] * qy + g[17] * qz + g[20];

  // coords = trunc((geom - (BX - DX/2)) / DX)   [trunc toward zero, as jnp.trunc]
  int xi = (int)truncf((gx + 54.0f) / 0.3f);
  int yi = (int)truncf((gy + 54.0f) / 0.3f);
  int zi = (int)truncf((gz + 10.0f) / 20.0f);
  if (xi < 0 || xi >= NX || yi < 0 || yi >= NY || zi < 0 || zi >= NZ) return;

  int b = bn / Nn;
  size_t pix = (size_t)h * FW + w;
  float dep = depth[((size_t)bn * D + dd) * HW + pix];
  const float* feat = dn3 + ((size_t)bn * Ctot + Coff) * HW + pix;
  float* dst = bev + ((((size_t)b * NZ + zi) * C) * NX + xi) * NY + yi;
  size_t chStride = (size_t)NX * NY;
  for (int c = 0; c < C; ++c) {
    float v = dep * feat[(size_t)c * HW];
    (void)__hip_atomic_fetch_add(dst + (size_t)c * chStride, v,
                                 __ATOMIC_RELAXED, __HIP_MEMORY_SCOPE_AGENT);
  }
}

// ---------------------------------------------------------------- launcher
static inline unsigned cdiv(long a, long b) { return (unsigned)((a + b - 1) / b); }

extern "C" void kernel_launch(void* const* d_in, const int* in_sizes, int n_in,
                              void* d_out, int out_size, void* d_ws, size_t ws_size,
                              hipStream_t stream) {
  (void)in_sizes; (void)n_in; (void)out_size; (void)ws_size;
  const float* x           = (const float*)d_in[0];
  const float* dmap        = (const float*)d_in[1];
  const float* intrins     = (const float*)d_in[2];
  const float* post_rots   = (const float*)d_in[3];
  const float* post_trans  = (const float*)d_in[4];
  const float* c2l_rots    = (const float*)d_in[5];
  const float* c2l_trans   = (const float*)d_in[6];
  const float* extra_rots  = (const float*)d_in[7];
  const float* extra_trans = (const float*)d_in[8];
  const float* dt_w1 = (const float*)d_in[9];
  const float* dt_s1 = (const float*)d_in[10];
  const float* dt_b1 = (const float*)d_in[11];
  const float* dt_w2 = (const float*)d_in[12];
  const float* dt_s2 = (const float*)d_in[13];
  const float* dt_b2 = (const float*)d_in[14];
  const float* dt_w3 = (const float*)d_in[15];
  const float* dt_s3 = (const float*)d_in[16];
  const float* dt_b3 = (const float*)d_in[17];
  const float* dn_w1 = (const float*)d_in[18];
  const float* dn_s1 = (const float*)d_in[19];
  const float* dn_b1 = (const float*)d_in[20];
  const float* dn_w2 = (const float*)d_in[21];
  const float* dn_s2 = (const float*)d_in[22];
  const float* dn_b2 = (const float*)d_in[23];
  const float* dn_w3 = (const float*)d_in[24];
  const float* dn_bias3 = (const float*)d_in[25];
  const float* ds_w1 = (const float*)d_in[26];
  const float* ds_s1 = (const float*)d_in[27];
  const float* ds_b1 = (const float*)d_in[28];
  const float* ds_w2 = (const float*)d_in[29];
  const float* ds_s2 = (const float*)d_in[30];
  const float* ds_b2 = (const float*)d_in[31];
  const float* ds_w3 = (const float*)d_in[32];
  const float* ds_s3 = (const float*)d_in[33];
  const float* ds_b3 = (const float*)d_in[34];

  float* ws    = (float*)d_ws;
  float* dt1   = ws + OFF_DT1;
  float* dt2   = ws + OFF_DT2;
  float* conc  = ws + OFF_CONCAT;
  float* dn1   = ws + OFF_DN1;
  float* dn2   = ws + OFF_DN2;
  float* dn3   = ws + OFF_DN3;
  float* depth = ws + OFF_DEPTH;
  float* geom  = ws + OFF_GEOM;
  float* bev   = ws + OFF_BEV;
  float* bev1  = ws + OFF_BEV1;
  float* bev2  = ws + OFF_BEV2;

  // ---- depth stem (tiny, direct conv) ----
  { long total = (long)BN6 * 8 * IH_ * IW_;
    conv_direct_kernel<<<cdiv(total, 256), 256, 0, stream>>>(
        dmap, dt_w1, dt_s1, dt_b1, dt1, BN6, 1, 8, IH_, IW_, IH_, IW_, 1, 1, 1, 0, 8, 0); }
  { long total = (long)BN6 * 32 * 64 * 176;
    conv_direct_kernel<<<cdiv(total, 256), 256, 0, stream>>>(
        dt1, dt_w2, dt_s2, dt_b2, dt2, BN6, 8, 32, IH_, IW_, 64, 176, 5, 5, 4, 2, 32, 0); }
  { long total = (long)BN6 * 64 * FH_ * FW_;   // writes channels [0,64) of concat
    conv_direct_kernel<<<cdiv(total, 256), 256, 0, stream>>>(
        dt2, dt_w3, dt_s3, dt_b3, conc, BN6, 32, 64, 64, 176, FH_, FW_, 5, 5, 2, 2, 320, 0); }
  { long total = (long)BN6 * INCH * HW_FEAT;   // x -> concat channels [64,320)
    copy_channels_kernel<<<cdiv(total, 256), 256, 0, stream>>>(
        x, conc, (int)total, INCH, HW_FEAT, 320, 64); }

  // ---- camera net (WMMA implicit GEMM, 16x64 tiles, LDS-staged weights) ----
  const int NPIX = BN6 * HW_FEAT;  // 16896
  { dim3 g(cdiv(NPIX, 64), 16); size_t sh = 16 * (320 + 2) * sizeof(float);
    conv_wmma_kernel<<<g, 32, sh, stream>>>(
        conc, dn_w1, dn_s1, dn_b1, dn1, BN6, 320, 256, FH_, FW_, FH_, FW_, 3, 3, 1, 1, 1); }
  { dim3 g(cdiv(NPIX, 64), 16); size_t sh = 16 * (256 + 2) * sizeof(float);
    conv_wmma_kernel<<<g, 32, sh, stream>>>(
        dn1, dn_w2, dn_s2, dn_b2, dn2, BN6, 256, 256, FH_, FW_, FH_, FW_, 3, 3, 1, 1, 1); }
  { dim3 g(cdiv(NPIX, 64), cdiv(DBINS + CCH, 16)); size_t sh = 16 * (256 + 2) * sizeof(float);
    conv_wmma_kernel<<<g, 32, sh, stream>>>(   // 1x1, bias only, no relu
        dn2, dn_w3, nullptr, dn_bias3, dn3, BN6, 256, DBINS + CCH, FH_, FW_, FH_, FW_, 1, 1, 1, 0, 0); }

  // ---- depth softmax ----
  { long total = (long)BN6 * HW_FEAT;
    softmax_depth_kernel<<<cdiv(total, 256), 256, 0, stream>>>(
        dn3, depth, BN6, DBINS, HW_FEAT, DBINS + CCH); }

  // ---- geometry prep ----
  geom_prep_kernel<<<1, 32, 0, stream>>>(
      intrins, post_rots, post_trans, c2l_rots, c2l_trans, extra_rots, extra_trans, geom, BN6, NCAM);

  // ---- BEV pooling (zero + atomic scatter-sum) ----
  { long total = (long)CCH * NXG * NYG * NZG;
    zero_kernel<<<cdiv(total, 256), 256, 0, stream>>>(bev, (int)total); }
  { long total = (long)BN6 * DBINS * HW_FEAT;
    scatter_kernel<<<cdiv(total, 256), 256, 0, stream>>>(
        dn3, depth, geom, bev, BN6, NCAM, DBINS, FH_, FW_, DBINS + CCH, DBINS, CCH, NXG, NYG, NZG); }

  // ---- BEV encoder (WMMA) ----
  { dim3 g(cdiv((long)NXG * NYG, 64), cdiv(CCH, 16)); size_t sh = 16 * (CCH + 2) * sizeof(float);
    conv_wmma_kernel<<<g, 32, sh, stream>>>(
        bev, ds_w1, ds_s1, ds_b1, bev1, 1, CCH, CCH, NXG, NYG, NXG, NYG, 3, 3, 1, 1, 1); }
  { dim3 g(cdiv((long)180 * 180, 64), cdiv(CCH, 16)); size_t sh = 16 * (CCH + 2) * sizeof(float);
    conv_wmma_kernel<<<g, 32, sh, stream>>>(
        bev1, ds_w2, ds_s2, ds_b2, bev2, 1, CCH, CCH, NXG, NYG, 180, 180, 3, 3, 2, 1, 1); }
  { dim3 g(cdiv((long)180 * 180, 64), cdiv(CCH, 16)); size_t sh = 16 * (CCH + 2) * sizeof(float);
    conv_wmma_kernel<<<g, 32, sh, stream>>>(
        bev2, ds_w3, ds_s3, ds_b3, (float*)d_out, 1, CCH, CCH, 180, 180, 180, 180, 3, 3, 1, 1, 1); }
}